// MSDSNv3_44667659878449
// MI455X (gfx1250) — compile-verified
//
#include <hip/hip_runtime.h>
#include <math.h>

typedef float v2f __attribute__((ext_vector_type(2)));
typedef float v8f __attribute__((ext_vector_type(8)));

#define HW 16384   // 128*128
#define NPIX 65536 // 4*128*128

// ---------------------------------------------------------------------------
// Stage 1a: pack block-sparse 1x1-conv weights into WkN[128][96] + bias96[96]
// Input-channel packing (K=128):
//   k in [0,16)        -> mask pconv output channels 0..15
//   k in [16+16g,32+16g)-> offset-group-g pconv output channels 0..15
//   k in [80,128)      -> x channels 16..63   (c = k - 64)
// Output-channel packing (N=96):
//   n in [0,9)   : mask taps (sigmoid applied later)
//   n = 9+18g+j  : offset group g, channel j (j = tap*2 + {dy,dx})
//   n in [81,96) : zero padding
// ---------------------------------------------------------------------------
__global__ void fill_w_kernel(const float* __restrict__ offw,   // (4,18,64)
                              const float* __restrict__ offb,   // (4,18)
                              const float* __restrict__ maskw,  // (9,64)
                              const float* __restrict__ maskb,  // (9,)
                              float* __restrict__ WkN,          // (128,96)
                              float* __restrict__ bias96)       // (96,)
{
    int tid = blockIdx.x * 256 + threadIdx.x;
    if (tid < 128 * 96) {
        int k = tid / 96, n = tid % 96;
        float v = 0.0f;
        if (n < 9) {
            if (k < 16)       v = maskw[n * 64 + k];
            else if (k >= 80) v = maskw[n * 64 + (k - 64)];
        } else if (n < 81) {
            int g = (n - 9) / 18, j = (n - 9) % 18;
            int klo = 16 + 16 * g;
            if (k >= klo && k < klo + 16) v = offw[(g * 18 + j) * 64 + (k - klo)];
            else if (k >= 80)             v = offw[(g * 18 + j) * 64 + (k - 64)];
        }
        WkN[k * 96 + n] = v;
    }
    int btid = tid - 128 * 96;
    if (btid >= 0 && btid < 96) {
        float v = 0.0f;
        if (btid < 9)       v = maskb[btid];
        else if (btid < 81) v = offb[btid - 9];   // flat (g*18+j) == n-9
        bias96[btid] = v;
    }
}

// ---------------------------------------------------------------------------
// Stage 1b: five 16->16 3x3 convs (pad 1) on x[:, :16].
// Z layout: [(conv*4 + b)*16 + oc][h][w], conv 0..3 = offset groups, 4 = mask.
// ---------------------------------------------------------------------------
__global__ void pconv3x3_kernel(const float* __restrict__ x,
                                const float* __restrict__ offpw,  // (4,16,16,3,3)
                                const float* __restrict__ maskpw, // (16,16,3,3)
                                float* __restrict__ Z)
{
    int blk = blockIdx.x;
    int pixblk = blk & 63;            // 64 pixel blocks of 256 per (conv,b,oc)
    int rest = blk >> 6;
    int oc = rest & 15; rest >>= 4;
    int b = rest & 3;
    int conv = rest >> 2;             // 0..4

    __shared__ float wsm[144];
    const float* wp = (conv < 4) ? (offpw + conv * 16 * 144) : maskpw;
    if (threadIdx.x < 144) wsm[threadIdx.x] = wp[oc * 144 + threadIdx.x];
    __syncthreads();

    int pix = pixblk * 256 + threadIdx.x;
    int h = pix >> 7, w = pix & 127;
    const float* xb = x + (size_t)b * 64 * HW;

    float acc = 0.0f;
#pragma unroll
    for (int ic = 0; ic < 16; ++ic) {
        const float* xp = xb + ic * HW;
#pragma unroll
        for (int ky = 0; ky < 3; ++ky) {
            int yy = h + ky - 1;
            if (yy < 0 || yy > 127) continue;
#pragma unroll
            for (int kx = 0; kx < 3; ++kx) {
                int xx = w + kx - 1;
                if (xx < 0 || xx > 127) continue;
                acc = fmaf(wsm[ic * 9 + ky * 3 + kx], xp[yy * 128 + xx], acc);
            }
        }
    }
    Z[(size_t)((conv * 4 + b) * 16 + oc) * HW + pix] = acc;
}

// ---------------------------------------------------------------------------
// Stage 2: WMMA f32 GEMM.  S[pixel][96] = sigmoid_or_id( A(65536x128) * W(128x96) + b )
// One wave computes one 16(M pixels) x 16(N channels) tile, K-loop step 4
// via V_WMMA_F32_16X16X4_F32 (32 WMMA ops per tile).
// A elements come straight from Z (k<80) or x channels 16..63 (k>=80).
// ---------------------------------------------------------------------------
__global__ void gemm_wmma_kernel(const float* __restrict__ Z,
                                 const float* __restrict__ x,
                                 const float* __restrict__ WkN,
                                 const float* __restrict__ bias96,
                                 float* __restrict__ S)
{
    int wave = (blockIdx.x * blockDim.x + threadIdx.x) >> 5;  // 0..24575
    int lane = threadIdx.x & 31;
    int mtile = wave & 4095;          // 4096 M tiles
    int ntile = wave >> 12;           // 6 N tiles

    int l15   = lane & 15;
    int khalf = (lane >> 4) << 1;     // 0 for lanes 0-15, 2 for lanes 16-31
    int arow  = mtile * 16 + l15;     // pixel row for A
    int b     = arow >> 14;           // batch
    int hw    = arow & 16383;
    int ccol  = ntile * 16 + l15;     // output channel for B / C

    v8f acc = {0.f, 0.f, 0.f, 0.f, 0.f, 0.f, 0.f, 0.f};

    for (int k0 = 0; k0 < 128; k0 += 4) {
        int kb = k0 + khalf;          // this lane-half's K pair base
        const float* ap;
        if (kb < 80)
            ap = Z + (size_t)(((kb >> 4) * 4 + b) * 16 + (kb & 15)) * HW + hw;
        else
            ap = x + (size_t)(b * 64 + 16 + (kb - 80)) * HW + hw;
        v2f A;
        A.x = ap[0];
        A.y = ap[HW];                 // next K = next channel plane
        v2f B;
        B.x = WkN[kb * 96 + ccol];
        B.y = WkN[(kb + 1) * 96 + ccol];
        acc = __builtin_amdgcn_wmma_f32_16x16x4_f32(
            false, A, false, B, (short)0, acc, false, false);
    }

    float bia = bias96[ccol];
    int crow = mtile * 16 + ((lane >> 4) << 3);   // rows 0-7 / 8-15 halves
#pragma unroll
    for (int r = 0; r < 8; ++r) {
        float v = acc[r] + bia;
        if (ccol < 9) v = 1.0f / (1.0f + expf(-v));   // sigmoid on mask cols
        S[(size_t)(crow + r) * 96 + ccol] = v;
    }
}

// ---------------------------------------------------------------------------
// Stage 3: modulated deformable sampling. One thread per (b,g,tap,pixel);
// bilinear weights computed once, reused for 16 channels x 2 y-batches.
// Lanes = consecutive w -> coalesced output stores (302 MB, the roofline term).
// ---------------------------------------------------------------------------
__global__ void deform_kernel(const float* __restrict__ y,
                              const float* __restrict__ S,
                              float* __restrict__ out)
{
    int tid = blockIdx.x * 256 + threadIdx.x;
    int pix = tid & 16383;
    int rest = tid >> 14;
    int tap = rest % 9;
    rest /= 9;
    int g = rest & 3;
    int b = rest >> 2;

    int h = pix >> 7, w = pix & 127;
    const float* sp = S + (size_t)(b * HW + pix) * 96;
    float mk = sp[tap];
    float dy = sp[9 + g * 18 + tap * 2];
    float dx = sp[9 + g * 18 + tap * 2 + 1];

    int dil = 2 * g + 1;
    float py = (float)(h + (tap / 3 - 1) * dil) + dy;
    float px = (float)(w + (tap % 3 - 1) * dil) + dx;

    float y0f = floorf(py), x0f = floorf(px);
    float wy = py - y0f, wx = px - x0f;
    int iy0 = (int)y0f, ix0 = (int)x0f;
    int iy1 = iy0 + 1, ix1 = ix0 + 1;
    float vy0 = (iy0 >= 0 && iy0 <= 127) ? 1.f : 0.f;
    float vy1 = (iy1 >= 0 && iy1 <= 127) ? 1.f : 0.f;
    float vx0 = (ix0 >= 0 && ix0 <= 127) ? 1.f : 0.f;
    float vx1 = (ix1 >= 0 && ix1 <= 127) ? 1.f : 0.f;
    int cy0 = min(max(iy0, 0), 127), cy1 = min(max(iy1, 0), 127);
    int cx0 = min(max(ix0, 0), 127), cx1 = min(max(ix1, 0), 127);
    float w00 = (1.f - wy) * (1.f - wx) * vy0 * vx0;
    float w01 = (1.f - wy) * wx * vy0 * vx1;
    float w10 = wy * (1.f - wx) * vy1 * vx0;
    float w11 = wy * wx * vy1 * vx1;
    int i00 = cy0 * 128 + cx0, i01 = cy0 * 128 + cx1;
    int i10 = cy1 * 128 + cx0, i11 = cy1 * 128 + cx1;

    const float* yp1 = y + (size_t)(b * 64 + g * 16) * HW;
    const float* yp2 = y + (size_t)((b + 4) * 64 + g * 16) * HW;
    float* op1 = out + (size_t)((b * 64 + g * 16) * 9 + tap) * HW + pix;
    float* op2 = out + (size_t)(((b + 4) * 64 + g * 16) * 9 + tap) * HW + pix;

#pragma unroll
    for (int c = 0; c < 16; ++c) {
        const float* p1 = yp1 + (size_t)c * HW;
        float v1 = w00 * p1[i00] + w01 * p1[i01] + w10 * p1[i10] + w11 * p1[i11];
        op1[(size_t)c * 9 * HW] = v1 * mk;
        const float* p2 = yp2 + (size_t)c * HW;
        float v2 = w00 * p2[i00] + w01 * p2[i01] + w10 * p2[i10] + w11 * p2[i11];
        op2[(size_t)c * 9 * HW] = v2 * mk;
    }
}

// ---------------------------------------------------------------------------
extern "C" void kernel_launch(void* const* d_in, const int* in_sizes, int n_in,
                              void* d_out, int out_size, void* d_ws, size_t ws_size,
                              hipStream_t stream) {
    const float* x      = (const float*)d_in[0];  // (4,64,128,128)
    const float* y      = (const float*)d_in[1];  // (8,64,128,128)
    const float* offpw  = (const float*)d_in[2];  // (4,16,16,3,3)
    const float* offw   = (const float*)d_in[3];  // (4,18,64)
    const float* offb   = (const float*)d_in[4];  // (4,18)
    const float* maskpw = (const float*)d_in[5];  // (16,16,3,3)
    const float* maskw  = (const float*)d_in[6];  // (9,64)
    const float* maskb  = (const float*)d_in[7];  // (9,)
    float* out = (float*)d_out;

    char* ws = (char*)d_ws;
    const size_t Z_BYTES = (size_t)5 * 4 * 16 * HW * 4;       // 20.97 MB
    const size_t S_BYTES = (size_t)NPIX * 96 * 4;             // 25.17 MB
    float* Z      = (float*)(ws);
    float* S      = (float*)(ws + Z_BYTES);
    float* WkN    = (float*)(ws + Z_BYTES + S_BYTES);
    float* bias96 = (float*)(ws + Z_BYTES + S_BYTES + 128 * 96 * 4);

    // Stage 1: weight packing (49 blocks) + five 3x3 convs (5*4*16*64 blocks)
    fill_w_kernel<<<49, 256, 0, stream>>>(offw, offb, maskw, maskb, WkN, bias96);
    pconv3x3_kernel<<<20480, 256, 0, stream>>>(x, offpw, maskpw, Z);

    // Stage 2: WMMA GEMM -> offsets + sigmoided mask, S[pixel][96]
    // 4096 M-tiles * 6 N-tiles = 24576 waves, 8 waves/block -> 3072 blocks
    gemm_wmma_kernel<<<3072, 256, 0, stream>>>(Z, x, WkN, bias96, S);

    // Stage 3: deformable sampling, 4*4*9*16384 threads -> 9216 blocks
    deform_kernel<<<9216, 256, 0, stream>>>(y, S, out);
}